// MyLoss_78099685310900
// MI455X (gfx1250) — compile-verified
//
#include <hip/hip_runtime.h>

// Problem constants from the reference: B=16, C=4, D=H=W=96.
#define N_SPATIAL      884736   // 96*96*96
#define N_PAIRS        64       // B*C
#define BLOCKS_PER_PAIR 24
#define CHUNK          36864    // N_SPATIAL / BLOCKS_PER_PAIR
#define WAVES_PER_BLK  8        // 256 threads / wave32
#define ELEMS_PER_WAVE_ITER 128 // 32 lanes * float4
#define ITERS          36       // CHUNK / (WAVES_PER_BLK * 128)

typedef __attribute__((ext_vector_type(2))) float v2f;
typedef __attribute__((ext_vector_type(8))) float v8f;

// Pass 1: each block reduces a 36864-element chunk of one (b,c) pair into
// (sum((in-lab)^2), sum(|lab|)) using V_WMMA_F32_16X16X4_F32 with an all-ones
// B matrix as a full-precision f32 matrix-engine accumulator.
__global__ __launch_bounds__(256) void mse_reduce_kernel(
    const float* __restrict__ in, const float* __restrict__ lab,
    float2* __restrict__ partials) {
  const int pair = blockIdx.x / BLOCKS_PER_PAIR;
  const int bip  = blockIdx.x % BLOCKS_PER_PAIR;
  const int wave = threadIdx.x >> 5;
  const int lane = threadIdx.x & 31;

  const long base  = (long)pair * N_SPATIAL + (long)bip * CHUNK;
  const long base4 = base >> 2;  // float4 index
  const float4* __restrict__ in4 = reinterpret_cast<const float4*>(in);
  const float4* __restrict__ lb4 = reinterpret_cast<const float4*>(lab);

  v8f cd = {};                 // accumulator for sum of squared diffs
  v8f ca = {};                 // accumulator for sum of |label|
  const v2f ones = {1.0f, 1.0f};  // B matrix (4x16 of ones) = 2 VGPRs/lane

  for (int j = 0; j < ITERS; ++j) {
    const long idx = base4 + (long)(j * WAVES_PER_BLK + wave) * 32 + lane;
    const float4 x = in4[idx];
    const float4 y = lb4[idx];

    const float dx = x.x - y.x, dy = x.y - y.y;
    const float dz = x.z - y.z, dw = x.w - y.w;

    v2f ad; ad[0] = dx * dx + dz * dz; ad[1] = dy * dy + dw * dw;
    v2f aa; aa[0] = fabsf(y.x) + fabsf(y.z); aa[1] = fabsf(y.y) + fabsf(y.w);

    // D = A x ones + C : accumulates per-row sums into every column of C.
    cd = __builtin_amdgcn_wmma_f32_16x16x4_f32(
        false, ad, false, ones, (short)0, cd, false, false);
    ca = __builtin_amdgcn_wmma_f32_16x16x4_f32(
        false, aa, false, ones, (short)0, ca, false, false);
  }

  // Column n of the 16x16 C holds all 16 row-sums split across lanes n and
  // n+16 (M=0..7 in lanes 0-15, M=8..15 in lanes 16-31). Fold within lane,
  // then across the half-wave.
  float sd = cd[0] + cd[1] + cd[2] + cd[3] + cd[4] + cd[5] + cd[6] + cd[7];
  float sa = ca[0] + ca[1] + ca[2] + ca[3] + ca[4] + ca[5] + ca[6] + ca[7];
  sd += __shfl_xor(sd, 16);
  sa += __shfl_xor(sa, 16);

  __shared__ float rd[WAVES_PER_BLK];
  __shared__ float ra[WAVES_PER_BLK];
  if (lane == 0) { rd[wave] = sd; ra[wave] = sa; }
  __syncthreads();
  if (threadIdx.x == 0) {
    float td = 0.0f, ta = 0.0f;
    for (int w = 0; w < WAVES_PER_BLK; ++w) { td += rd[w]; ta += ra[w]; }
    partials[blockIdx.x] = make_float2(td, ta);  // private slot: no race, no zeroing needed
  }
}

// Pass 2: 64 threads, one per (b,c) pair.
// loss = (1/B) * sum_{b,c} [ sum_d2 / (C * sum_absl) ]   (spatial N cancels)
__global__ __launch_bounds__(64) void mse_combine_kernel(
    const float2* __restrict__ partials, float* __restrict__ out) {
  const int t = threadIdx.x;  // pair index 0..63
  float d2 = 0.0f, al = 0.0f;
  for (int i = 0; i < BLOCKS_PER_PAIR; ++i) {
    const float2 p = partials[t * BLOCKS_PER_PAIR + i];
    d2 += p.x; al += p.y;
  }
  __shared__ float sb[N_PAIRS];
  sb[t] = d2 / (4.0f * al);   // C = 4
  __syncthreads();
  if (t == 0) {
    float tot = 0.0f;
    for (int i = 0; i < N_PAIRS; ++i) tot += sb[i];
    out[0] = tot * (1.0f / 16.0f);  // / B
  }
}

extern "C" void kernel_launch(void* const* d_in, const int* in_sizes, int n_in,
                              void* d_out, int out_size, void* d_ws, size_t ws_size,
                              hipStream_t stream) {
  const float* in  = (const float*)d_in[0];
  const float* lab = (const float*)d_in[1];
  float2* partials = (float2*)d_ws;  // 1536 float2 = 12 KB
  float* out       = (float*)d_out;

  mse_reduce_kernel<<<N_PAIRS * BLOCKS_PER_PAIR, 256, 0, stream>>>(in, lab, partials);
  mse_combine_kernel<<<1, N_PAIRS, 0, stream>>>(partials, out);
}